// Model_34703335752059
// MI455X (gfx1250) — compile-verified
//
#include <hip/hip_runtime.h>
#include <cstdint>
#include <cstddef>

// out[b,d] = silu( cs[idx[b],d,0]*W[d,0] + cs[.,d,1]*W[d,1] + cs[.,d,2]*W[d,2]
//                  + x[b,d]*W[d,3] + bias[d] )
// B=2048, D=4096, width=4. Pure streaming kernel, HBM-bound (~160MB traffic).

typedef __attribute__((ext_vector_type(4))) float f4;
typedef __attribute__((address_space(3))) float lds_float;

#define BATCH_N 2048
#define DIM_N   4096
#define CPB     1024   // channels per block
#define NTHR    256    // 8 waves (wave32)

__global__ __launch_bounds__(NTHR) void dwconv_silu_kernel(
    const float* __restrict__ x,
    const float* __restrict__ conv_state,
    const float* __restrict__ weight,
    const float* __restrict__ bias,
    const int*   __restrict__ conv_idx,
    float* __restrict__ out)
{
    // 12 KB tile: conv_state[row, cbase..cbase+1023, 0..2]
    __shared__ __align__(16) float tile[CPB * 3];

    const int b     = blockIdx.y;
    const int cbase = blockIdx.x * CPB;
    const int t     = threadIdx.x;
    const int wave  = t >> 5;   // 0..7
    const int lane  = t & 31;

    // Uniform per block: gather row of conv_state
    const int row = conv_idx[b];
    const float* srow =
        conv_state + (size_t)row * (size_t)(DIM_N * 3) + (size_t)cbase * 3;

    // LDS byte offset of the tile (addrspace(3) pointers are 32-bit LDS offsets)
    const unsigned lds_base =
        (unsigned)(unsigned long long)(lds_float*)tile;

    // Wave-self-contained staging: wave w fills float4 slots [96w, 96w+96),
    // lane l issues slots {96w+l, 96w+32+l, 96w+64+l}. The per-instruction
    // offset is added to BOTH the LDS and global addresses by the hardware.
    const unsigned byteoff = (unsigned)(96 * wave + lane) * 16u;
    const unsigned ldsaddr = lds_base + byteoff;

    asm volatile("global_load_async_to_lds_b128 %0, %1, %2 offset:0"
                 :: "v"(ldsaddr), "v"(byteoff), "s"(srow) : "memory");
    asm volatile("global_load_async_to_lds_b128 %0, %1, %2 offset:512"
                 :: "v"(ldsaddr), "v"(byteoff), "s"(srow) : "memory");
    asm volatile("global_load_async_to_lds_b128 %0, %1, %2 offset:1024"
                 :: "v"(ldsaddr), "v"(byteoff), "s"(srow) : "memory");

    // Overlap: direct 128-bit loads of x / weight / bias while the async
    // engine streams conv_state into LDS. weight+bias (80KB total) stay in L2.
    const int g4 = (cbase >> 2) + t;                      // float4 column index
    const f4 xv = ((const f4*)x)[(size_t)b * (DIM_N / 4) + g4];
    const f4 bv = ((const f4*)bias)[g4];
    const int ch0 = cbase + 4 * t;                        // first of 4 channels
    f4 wj[4];
    wj[0] = ((const f4*)weight)[ch0 + 0];
    wj[1] = ((const f4*)weight)[ch0 + 1];
    wj[2] = ((const f4*)weight)[ch0 + 2];
    wj[3] = ((const f4*)weight)[ch0 + 3];

    // Wait for this wave's async copies; reads below touch only this wave's
    // slots, so no workgroup barrier is required.
    asm volatile("s_wait_asynccnt 0" ::: "memory");

    // Thread t handles tile channels 4t..4t+3 -> floats 12t..12t+11
    const f4* tile4 = (const f4*)tile;
    const f4 c0 = tile4[3 * t + 0];  // d0w0 d0w1 d0w2 d1w0
    const f4 c1 = tile4[3 * t + 1];  // d1w1 d1w2 d2w0 d2w1
    const f4 c2 = tile4[3 * t + 2];  // d2w2 d3w0 d3w1 d3w2

    const float cs0[4] = {c0[0], c0[3], c1[2], c2[1]};
    const float cs1[4] = {c0[1], c1[0], c1[3], c2[2]};
    const float cs2[4] = {c0[2], c1[1], c2[0], c2[3]};

    f4 o;
    #pragma unroll
    for (int j = 0; j < 4; ++j) {
        float a = cs0[j] * wj[j][0] + cs1[j] * wj[j][1] + cs2[j] * wj[j][2]
                + xv[j]  * wj[j][3] + bv[j];
        // silu(a) = a / (1 + e^-a); e^-a = 2^(-a*log2(e))
        float e = __builtin_amdgcn_exp2f(a * -1.4426950408889634f);
        o[j] = a * __builtin_amdgcn_rcpf(1.0f + e);
    }

    ((f4*)out)[(size_t)b * (DIM_N / 4) + g4] = o;
}

extern "C" void kernel_launch(void* const* d_in, const int* in_sizes, int n_in,
                              void* d_out, int out_size, void* d_ws, size_t ws_size,
                              hipStream_t stream) {
    const float* x    = (const float*)d_in[0];  // (2048, 4096)
    const float* cs   = (const float*)d_in[1];  // (2048, 4096, 3)
    const float* w    = (const float*)d_in[2];  // (4096, 4)
    const float* bias = (const float*)d_in[3];  // (4096,)
    const int*   idx  = (const int*)d_in[4];    // (2048,)
    float* out = (float*)d_out;                 // (2048, 4096)

    dim3 grid(DIM_N / CPB, BATCH_N);            // (4, 2048) blocks
    dwconv_silu_kernel<<<grid, NTHR, 0, stream>>>(x, cs, w, bias, idx, out);
}